// curveModel_22144851378509
// MI455X (gfx1250) — compile-verified
//
#include <hip/hip_runtime.h>
#include <hip/hip_bf16.h>

typedef float v2f __attribute__((ext_vector_type(2)));
typedef float v8f __attribute__((ext_vector_type(8)));

#define NTOT   4194304   // curve segments
#define NB     4096      // blocks per curve
#define CHUNK  1024      // elements per block
#define TILE   256       // elements per WMMA tile (16x16)
#define TPB    4         // tiles per block  (TPB*TILE == CHUNK)
#define NTS    (NB / TILE)  // tiles in the cross-block scan

__device__ __forceinline__ v8f wmma4(v2f a, v2f b, v8f c) {
  // D(16x16) = A(16x4) * B(4x16) + C
  return __builtin_amdgcn_wmma_f32_16x16x4_f32(false, a, false, b, (short)0, c,
                                               false, false);
}

// Inclusive cumsum of 256 f32 staged in lds (element e = M*16+N), result in
// WMMA C/D layout: lane L, slot r holds element e = (r + (L>=16)*8)*16 + (L&15).
// runOff is added to every element; grand = total of all 256 (uniform).
__device__ __forceinline__ void tile_cumsum(const float* lds, int lane,
                                            const v2f* bU, float runOff,
                                            v8f& outVal, float& grand) {
  const int n  = lane & 15;
  const int cb = (lane < 16) ? 0 : 2;
  v2f A[4];
#pragma unroll
  for (int k = 0; k < 4; ++k)
    A[k] = *(const v2f*)(lds + n * 16 + 4 * k + cb);   // X[M=n][K=4k+cb(+1)]

  v8f Y = {0.f, 0.f, 0.f, 0.f, 0.f, 0.f, 0.f, 0.f};
  v8f T = {0.f, 0.f, 0.f, 0.f, 0.f, 0.f, 0.f, 0.f};
#pragma unroll
  for (int k = 0; k < 4; ++k) Y = wmma4(A[k], bU[k], Y);   // X*U: per-row cumsum
  v2f ones; ones.x = 1.f; ones.y = 1.f;
#pragma unroll
  for (int k = 0; k < 4; ++k) T = wmma4(A[k], ones, T);    // X*J: T[M][*]=rowTotal_M

  float excl[8];
  excl[0] = 0.f;
#pragma unroll
  for (int r = 1; r < 8; ++r) excl[r] = excl[r - 1] + T[r - 1];
  float halfSum = excl[7] + T[7];                 // h0: G0, h1: G1
  float other   = __shfl_xor(halfSum, 16, 32);    // cross-half swap
  float base    = (lane >= 16) ? other : 0.f;     // rows 8..15 add G0
  grand = halfSum + other;
#pragma unroll
  for (int r = 0; r < 8; ++r) outVal[r] = Y[r] + (runOff + base + excl[r]);
}

__device__ __forceinline__ void make_bU(int lane, v2f* bU) {
  const int n = lane & 15;
  const int cb = (lane < 16) ? 0 : 2;
#pragma unroll
  for (int k = 0; k < 4; ++k) {
    int K0 = 4 * k + cb;
    bU[k].x = (K0 <= n) ? 1.f : 0.f;
    bU[k].y = (K0 + 1 <= n) ? 1.f : 0.f;
  }
}

template <bool WRITE>
__global__ __launch_bounds__(32)
void curve_blocks(const float* __restrict__ vecA, const float* __restrict__ vecB,
                  const float* __restrict__ dlp,
                  const float* __restrict__ offAng, const float* __restrict__ offX,
                  const float* __restrict__ offYv,
                  const float* __restrict__ PS, const float* __restrict__ PE,
                  float* __restrict__ blk, float* __restrict__ out) {
  __shared__ __align__(16) float sAng[TILE];
  __shared__ __align__(16) float sSx[TILE];
  __shared__ __align__(16) float sSy[TILE];

  const int lane = threadIdx.x;           // one wave32 per block
  const int b = blockIdx.x;
  const int c = blockIdx.y;               // curve index
  const float* vec = c ? vecB : vecA;
  const float dl = dlp[0];

  float runAngle, runX, runY;
  if (WRITE) {
    runAngle = offAng[c * NB + b];
    runX = offX[c * NB + b];
    runY = offYv[c * NB + b];
  } else {
    runAngle = 0.f; runX = 0.f; runY = 0.f;   // local frame; rotated in scan pass
  }

  const int n = lane & 15;
  const int rowOff = (lane < 16) ? 0 : 8;
  v2f bU[4];
  make_bU(lane, bU);

  const size_t chunkBase = (size_t)b * CHUNK;
  float* outC = WRITE ? (out + (size_t)c * 2u * (size_t)(NTOT + 1)) : nullptr;

  // register double-buffer of the staged global tile
  const float4* gv0 = (const float4*)(vec + chunkBase);
  float4 c0 = gv0[lane * 2 + 0];
  float4 c1 = gv0[lane * 2 + 1];

  for (int t = 0; t < TPB; ++t) {
    *(float4*)&sAng[lane * 8 + 0] = c0;
    *(float4*)&sAng[lane * 8 + 4] = c1;
    if (t + 1 < TPB) {                    // issue next tile's loads early
      const float4* gn = (const float4*)(vec + chunkBase + (size_t)(t + 1) * TILE);
      c0 = gn[lane * 2 + 0];
      c1 = gn[lane * 2 + 1];
    }
    __syncthreads();

    v8f theta; float gA;
    tile_cumsum(sAng, lane, bU, runAngle, theta, gA);
    runAngle += gA;

    v8f sx, sy;
#pragma unroll
    for (int r = 0; r < 8; ++r) {
      float s, cc;
      sincosf(theta[r], &s, &cc);         // shared argument reduction
      sx[r] = dl * cc;
      sy[r] = dl * s;
    }
    __syncthreads();
#pragma unroll
    for (int r = 0; r < 8; ++r) {         // D-layout -> LDS (e-order), conflict-free
      int e = (r + rowOff) * 16 + n;
      sSx[e] = sx[r];
      sSy[e] = sy[r];
    }
    __syncthreads();

    v8f px, py; float gX, gY;
    tile_cumsum(sSx, lane, bU, runX, px, gX);
    tile_cumsum(sSy, lane, bU, runY, py, gY);
    runX += gX; runY += gY;

    if (WRITE) {
      const size_t base = chunkBase + (size_t)t * TILE;
#pragma unroll
      for (int r = 0; r < 8; ++r) {
        size_t e = (size_t)(r + rowOff) * 16 + (size_t)n;
        size_t g = base + e + 1;          // output row (row 0 = start point)
        v2f p; p.x = px[r]; p.y = py[r];
        *(v2f*)(outC + 2 * g) = p;        // 8B-aligned coalesced pair store
      }
    }
    __syncthreads();
  }

  if (!WRITE) {
    if (lane == 0) {                      // block totals (uniform across lanes)
      blk[(0 * 2 + c) * NB + b] = runAngle;  // A_b
      blk[(1 * 2 + c) * NB + b] = runX;      // Sx_b (local frame)
      blk[(2 * 2 + c) * NB + b] = runY;      // Sy_b (local frame)
    }
  } else if (b == 0 && lane == 0) {
    const float* st = c ? PE : PS;
    outC[0] = st[0];
    outC[1] = st[1];
  }
}

// Cross-block scan, WMMA-parallel (one wave per curve):
//   offAng_b = the0 + sum_{k<b} A_k
//   offPos_b = start + sum_{k<b} R(offAng_k) * S_k
__global__ __launch_bounds__(32)
void scan_blocks(const float* __restrict__ blk,
                 const float* __restrict__ the0p, const float* __restrict__ the02p,
                 const float* __restrict__ PS, const float* __restrict__ PE,
                 float* __restrict__ offAng, float* __restrict__ offX,
                 float* __restrict__ offY) {
  __shared__ __align__(16) float sA[TILE];
  __shared__ __align__(16) float sDx[TILE];
  __shared__ __align__(16) float sDy[TILE];

  const int lane = threadIdx.x;
  const int c = blockIdx.x;
  const float* A  = blk + (0 * 2 + c) * NB;
  const float* Sx = blk + (1 * 2 + c) * NB;
  const float* Sy = blk + (2 * 2 + c) * NB;
  const float th0 = c ? the02p[0] : the0p[0];
  const float* st = c ? PE : PS;
  float runAng = th0;
  float runX = st[0], runY = st[1];

  const int n = lane & 15;
  const int rowOff = (lane < 16) ? 0 : 8;
  v2f bU[4];
  make_bU(lane, bU);

  if (lane == 0) {                        // exclusive scan: first entries
    offAng[c * NB] = th0;
    offX[c * NB]   = runX;
    offY[c * NB]   = runY;
  }

  for (int t = 0; t < NTS; ++t) {
    const int base = t * TILE;
    const float4* gA4 = (const float4*)(A + base);
    float4 a0 = gA4[lane * 2 + 0];
    float4 a1 = gA4[lane * 2 + 1];
    *(float4*)&sA[lane * 8 + 0] = a0;
    *(float4*)&sA[lane * 8 + 4] = a1;
    __syncthreads();

    v8f incA; float gAng;
    tile_cumsum(sA, lane, bU, runAng, incA, gAng);

    // originals in D layout (A from LDS, S directly from global, coalesced)
    v8f dx, dy;
#pragma unroll
    for (int r = 0; r < 8; ++r) {
      int e = (r + rowOff) * 16 + n;
      float av  = sA[e];
      float sxv = Sx[base + e];
      float syv = Sy[base + e];
      float phi = incA[r] - av;           // exclusive angle = offAng at entry
      float sa, ca;
      sincosf(phi, &sa, &ca);
      dx[r] = ca * sxv - sa * syv;        // rotate local displacement
      dy[r] = sa * sxv + ca * syv;
    }
    __syncthreads();
#pragma unroll
    for (int r = 0; r < 8; ++r) {
      int e = (r + rowOff) * 16 + n;
      sDx[e] = dx[r];
      sDy[e] = dy[r];
    }
    __syncthreads();

    v8f ix, iy; float gX, gY;
    tile_cumsum(sDx, lane, bU, runX, ix, gX);
    tile_cumsum(sDy, lane, bU, runY, iy, gY);

    // scatter inclusive values to entry b+1 (exclusive-scan outputs)
#pragma unroll
    for (int r = 0; r < 8; ++r) {
      int g = base + (r + rowOff) * 16 + n;
      int idx = g + 1;
      if (idx < NB) {
        offAng[c * NB + idx] = incA[r];
        offX[c * NB + idx]   = ix[r];
        offY[c * NB + idx]   = iy[r];
      }
    }
    runAng += gAng; runX += gX; runY += gY;
    __syncthreads();
  }
}

extern "C" void kernel_launch(void* const* d_in, const int* in_sizes, int n_in,
                              void* d_out, int out_size, void* d_ws, size_t ws_size,
                              hipStream_t stream) {
  (void)in_sizes; (void)n_in; (void)out_size; (void)ws_size;
  const float* vec   = (const float*)d_in[0];
  const float* vec2  = (const float*)d_in[1];
  const float* the0  = (const float*)d_in[2];
  const float* the02 = (const float*)d_in[3];
  const float* PS    = (const float*)d_in[4];
  const float* PE    = (const float*)d_in[5];
  const float* dl    = (const float*)d_in[6];
  float* out = (float*)d_out;

  float* ws     = (float*)d_ws;         // 192 KB of scratch used
  float* blk    = ws;                   // [3][2][NB]
  float* offAng = ws + 3 * 2 * NB;      // [2][NB]
  float* offX   = offAng + 2 * NB;      // [2][NB]
  float* offY   = offX + 2 * NB;        // [2][NB]

  dim3 grid(NB, 2);
  curve_blocks<false><<<grid, 32, 0, stream>>>(vec, vec2, dl,
                                               nullptr, nullptr, nullptr,
                                               PS, PE, blk, nullptr);
  scan_blocks<<<2, 32, 0, stream>>>(blk, the0, the02, PS, PE, offAng, offX, offY);
  curve_blocks<true><<<grid, 32, 0, stream>>>(vec, vec2, dl,
                                              offAng, offX, offY,
                                              PS, PE, nullptr, out);
}